// Half_Graph_51539607552520
// MI455X (gfx1250) — compile-verified
//
#include <hip/hip_runtime.h>
#include <hip/hip_bf16.h>

typedef __attribute__((ext_vector_type(2))) float v2f;
typedef __attribute__((ext_vector_type(8))) float v8f;

#define HWSZ 16384   // H*W
#define CCH  256     // h_fea channels
#define NIMG 8

__device__ __forceinline__ float sigmoidf(float x) { return 1.0f / (1.0f + __expf(-x)); }
__device__ __forceinline__ float reluf(float x)    { return fmaxf(x, 0.0f); }

// One workgroup = 128 threads = 4 waves; each wave handles 32 consecutive pixels.
// WMMA GEMM: D[pixel(16) x out(16)] += A[pixel x K=4] * B[K=4 x out], K-loop over 256 h_fea channels.
// Packed output columns: 0..19 = pdp_w1 rows, 20 = decU_att_w, 21 = decL_att_w, 22..31 = zero pad.
__global__ __launch_bounds__(128)
void half_graph_fused(
    const float* __restrict__ h_fea, const float* __restrict__ xh_u, const float* __restrict__ xh_l,
    const float* __restrict__ xf,    const float* __restrict__ xp,
    const float* __restrict__ pdp_w1, const float* __restrict__ pdp_w2,
    const float* __restrict__ att_w,  const float* __restrict__ att_b,
    const float* __restrict__ compU_att_w, const float* __restrict__ compU_att_b, const float* __restrict__ compU_w,
    const float* __restrict__ compL_att_w, const float* __restrict__ compL_att_b, const float* __restrict__ compL_w,
    const float* __restrict__ decU_att_w,  const float* __restrict__ decU_att_b,  const float* __restrict__ decU_w,
    const float* __restrict__ decL_att_w,  const float* __restrict__ decL_att_b,  const float* __restrict__ decL_w,
    const float* __restrict__ updU_gate_w, const float* __restrict__ updU_gate_b, const float* __restrict__ updU_cand_w,
    const float* __restrict__ updL_gate_w, const float* __restrict__ updL_gate_b, const float* __restrict__ updL_cand_w,
    float* __restrict__ out)
{
    // Packed weights as K-pairs: Wp[m][o] = {Wc[o][2m], Wc[o][2m+1]}, m in [0,128), o in [0,32)
    __shared__ float Wp[128 * 32 * 2];          // 32 KB
    __shared__ float TB[4][32 * 33];            // per-wave D transpose, stride 33 (bank-conflict free)

    const int tid   = threadIdx.x;
    const int wave  = tid >> 5;
    const int lane  = tid & 31;
    const int lhalf = lane >> 4;                // which K half-pair this lane supplies
    const int lmod  = lane & 15;

    // ---- stage packed big-weight matrix into LDS ----
    for (int e = tid; e < 128 * 32; e += 128) {
        int m = e >> 5, o = e & 31, k = m * 2;
        float w0 = 0.0f, w1 = 0.0f;
        if (o < 20)       { w0 = pdp_w1[o * 276 + k];  w1 = pdp_w1[o * 276 + k + 1]; }
        else if (o == 20) { w0 = decU_att_w[k];        w1 = decU_att_w[k + 1]; }
        else if (o == 21) { w0 = decL_att_w[k];        w1 = decL_att_w[k + 1]; }
        Wp[e * 2] = w0; Wp[e * 2 + 1] = w1;
    }
    __syncthreads();

    const int img   = blockIdx.x >> 7;                       // 128 WGs per image
    const int pbase = (blockIdx.x & 127) * 128 + wave * 32;  // 32 pixels per wave
    const float* hb = h_fea + (size_t)img * CCH * HWSZ;

    // ---- K-loop: 4 WMMA chains (2 pixel tiles x 2 out tiles), K=256 in steps of 4 ----
    v8f acc00 = {}, acc01 = {}, acc10 = {}, acc11 = {};
    for (int k0 = 0; k0 < CCH; k0 += 4) {
        const int c = k0 + 2 * lhalf;
        const float* pa = hb + (size_t)c * HWSZ + pbase + lmod;
        v2f a0, a1;                                   // A: lane half supplies K pair {c, c+1}
        a0.x = pa[0];   a0.y = pa[HWSZ];
        a1.x = pa[16];  a1.y = pa[HWSZ + 16];

        const int cpf = (c + 16) & (CCH - 1);         // prefetch 16 channels ahead (wraps, stays in-bounds)
        __builtin_prefetch(hb + (size_t)cpf * HWSZ + pbase + lmod, 0, 1);

        const int m = (k0 >> 1) + lhalf;              // B: K-pair row from LDS (ds_load_b64)
        v2f b0 = *(const v2f*)&Wp[(m * 32 + lmod) * 2];
        v2f b1 = *(const v2f*)&Wp[(m * 32 + 16 + lmod) * 2];

        acc00 = __builtin_amdgcn_wmma_f32_16x16x4_f32(false, a0, false, b0, (short)0, acc00, false, false);
        acc01 = __builtin_amdgcn_wmma_f32_16x16x4_f32(false, a0, false, b1, (short)0, acc01, false, false);
        acc10 = __builtin_amdgcn_wmma_f32_16x16x4_f32(false, a1, false, b0, (short)0, acc10, false, false);
        acc11 = __builtin_amdgcn_wmma_f32_16x16x4_f32(false, a1, false, b1, (short)0, acc11, false, false);
    }

    // ---- transpose D fragments so each lane owns one pixel's 22 projections ----
    float* tb = TB[wave];
    #pragma unroll
    for (int r = 0; r < 8; ++r) {
        const int pr = lhalf ? (8 + r) : r;           // D layout: lanes 16-31 hold M=8..15
        tb[pr * 33 + lmod]             = acc00[r];
        tb[pr * 33 + 16 + lmod]        = acc01[r];
        tb[(16 + pr) * 33 + lmod]      = acc10[r];
        tb[(16 + pr) * 33 + 16 + lmod] = acc11[r];
    }
    __syncthreads();

    // ================= epilogue: one pixel per lane =================
    const int p = pbase + lane;
    float z[22];
    #pragma unroll
    for (int o = 0; o < 22; ++o) z[o] = tb[lane * 33 + o];

    float xu[10], xl[10], xff[10];
    {
        const float* pu = xh_u + (size_t)img * 10 * HWSZ + p;
        const float* pl = xh_l + (size_t)img * 10 * HWSZ + p;
        const float* pf = xf   + (size_t)img * 10 * HWSZ + p;
        #pragma unroll
        for (int j = 0; j < 10; ++j) { xu[j] = pu[j * HWSZ]; xl[j] = pl[j * HWSZ]; xff[j] = pf[j * HWSZ]; }
    }

    // pdp: finish 276-ch projection (tail 20 channels), relu, then grouped 10x10 convs
    float t[20];
    #pragma unroll
    for (int o = 0; o < 20; ++o) {
        float s = z[o];
        #pragma unroll
        for (int j = 0; j < 10; ++j) s += xu[j] * pdp_w1[o * 276 + 256 + j];
        #pragma unroll
        for (int j = 0; j < 10; ++j) s += xl[j] * pdp_w1[o * 276 + 266 + j];
        t[o] = reluf(s);
    }
    float dpu[10], dpl[10];
    #pragma unroll
    for (int i = 0; i < 10; ++i) {
        float su = 0.0f, sl = 0.0f;
        #pragma unroll
        for (int j = 0; j < 10; ++j) {
            su += pdp_w2[i * 10 + j] * t[j];
            sl += pdp_w2[100 + i * 10 + j] * t[10 + j];
        }
        dpu[i] = reluf(su); dpl[i] = reluf(sl);
    }

    // grouped attention gates
    float agu, agl;
    {
        float su = att_b[0], sl = att_b[1];
        #pragma unroll
        for (int j = 0; j < 10; ++j) { su += att_w[j] * xu[j]; sl += att_w[10 + j] * xl[j]; }
        agu = sigmoidf(su); agl = sigmoidf(sl);
    }

    // decomp U/L: finish 276-ch att dot (tails), sigmoid; xfh = relu(att * (decW @ xf))
    float au = z[20] + decU_att_b[0];
    float al = z[21] + decL_att_b[0];
    #pragma unroll
    for (int j = 0; j < 10; ++j) {
        au += xff[j] * decU_att_w[256 + j] + xu[j] * decU_att_w[266 + j];
        al += xff[j] * decL_att_w[256 + j] + xl[j] * decL_att_w[266 + j];
    }
    au = sigmoidf(au); al = sigmoidf(al);
    float xfhu[10], xfhl[10];
    #pragma unroll
    for (int i = 0; i < 10; ++i) {
        float su = 0.0f, sl = 0.0f;
        #pragma unroll
        for (int j = 0; j < 10; ++j) { su += decU_w[i * 10 + j] * xff[j]; sl += decL_w[i * 10 + j] * xff[j]; }
        xfhu[i] = reluf(au * su); xfhl[i] = reluf(al * sl);
    }

    // comp U (4 parts) and comp L (2 parts)
    float msg[10], xphu[10], xphl[10];
    #pragma unroll
    for (int c = 0; c < 10; ++c) msg[c] = 0.0f;
    #pragma unroll
    for (int part = 0; part < 4; ++part) {
        const float* xpp = xp + ((size_t)(part * NIMG + img) * 10) * HWSZ + p;
        float v[10]; float s = compU_att_b[part];
        #pragma unroll
        for (int c = 0; c < 10; ++c) { v[c] = xpp[c * HWSZ]; s += compU_att_w[part * 10 + c] * v[c]; }
        s = sigmoidf(s);
        #pragma unroll
        for (int c = 0; c < 10; ++c) msg[c] += v[c] * s;
    }
    #pragma unroll
    for (int i = 0; i < 10; ++i) {
        float s = 0.0f;
        #pragma unroll
        for (int j = 0; j < 10; ++j) s += compU_w[i * 20 + j] * xu[j] + compU_w[i * 20 + 10 + j] * msg[j];
        xphu[i] = reluf(s);
    }
    #pragma unroll
    for (int c = 0; c < 10; ++c) msg[c] = 0.0f;
    #pragma unroll
    for (int part = 0; part < 2; ++part) {
        const float* xpp = xp + ((size_t)((part + 4) * NIMG + img) * 10) * HWSZ + p;
        float v[10]; float s = compL_att_b[part];
        #pragma unroll
        for (int c = 0; c < 10; ++c) { v[c] = xpp[c * HWSZ]; s += compL_att_w[part * 10 + c] * v[c]; }
        s = sigmoidf(s);
        #pragma unroll
        for (int c = 0; c < 10; ++c) msg[c] += v[c] * s;
    }
    #pragma unroll
    for (int i = 0; i < 10; ++i) {
        float s = 0.0f;
        #pragma unroll
        for (int j = 0; j < 10; ++j) s += compL_w[i * 20 + j] * xl[j] + compL_w[i * 20 + 10 + j] * msg[j];
        xphl[i] = reluf(s);
    }

    // message sums and GRU-style updates
    float mu[10], ml[10];
    #pragma unroll
    for (int i = 0; i < 10; ++i) {
        mu[i] = xphu[i] + (dpl[i] * agl + xu[i] * agu) + xfhu[i];
        ml[i] = xphl[i] + (dpu[i] * agu + xl[i] * agl) + xfhl[i];
    }

    float* outU = out + (size_t)img * 10 * HWSZ + p;
    float* outL = out + 1310720 + (size_t)img * 10 * HWSZ + p;
    #pragma unroll
    for (int i = 0; i < 10; ++i) {
        float gsU = updU_gate_b[i], csU = 0.0f, gsL = updL_gate_b[i], csL = 0.0f;
        #pragma unroll
        for (int j = 0; j < 10; ++j) {
            gsU += updU_gate_w[i * 20 + j] * xu[j] + updU_gate_w[i * 20 + 10 + j] * mu[j];
            csU += updU_cand_w[i * 20 + j] * xu[j] + updU_cand_w[i * 20 + 10 + j] * mu[j];
            gsL += updL_gate_w[i * 20 + j] * xl[j] + updL_gate_w[i * 20 + 10 + j] * ml[j];
            csL += updL_cand_w[i * 20 + j] * xl[j] + updL_cand_w[i * 20 + 10 + j] * ml[j];
        }
        const float gU = sigmoidf(gsU), gL = sigmoidf(gsL);
        outU[i * HWSZ] = xu[i] * (1.0f - gU) + reluf(csU) * gU;
        outL[i * HWSZ] = xl[i] * (1.0f - gL) + reluf(csL) * gL;
    }
    out[2621440 + (size_t)img * HWSZ + p] = au;
    out[2752512 + (size_t)img * HWSZ + p] = al;
}

extern "C" void kernel_launch(void* const* d_in, const int* in_sizes, int n_in,
                              void* d_out, int out_size, void* d_ws, size_t ws_size,
                              hipStream_t stream) {
    (void)in_sizes; (void)n_in; (void)out_size; (void)d_ws; (void)ws_size;
    const float* h_fea       = (const float*)d_in[0];
    const float* xh_u        = (const float*)d_in[1];
    const float* xh_l        = (const float*)d_in[2];
    const float* xf          = (const float*)d_in[3];
    const float* xp          = (const float*)d_in[4];
    const float* pdp_w1      = (const float*)d_in[5];
    const float* pdp_w2      = (const float*)d_in[6];
    const float* att_w       = (const float*)d_in[7];
    const float* att_b       = (const float*)d_in[8];
    const float* compU_att_w = (const float*)d_in[9];
    const float* compU_att_b = (const float*)d_in[10];
    const float* compU_w     = (const float*)d_in[11];
    const float* compL_att_w = (const float*)d_in[12];
    const float* compL_att_b = (const float*)d_in[13];
    const float* compL_w     = (const float*)d_in[14];
    const float* decU_att_w  = (const float*)d_in[15];
    const float* decU_att_b  = (const float*)d_in[16];
    const float* decU_w      = (const float*)d_in[17];
    const float* decL_att_w  = (const float*)d_in[18];
    const float* decL_att_b  = (const float*)d_in[19];
    const float* decL_w      = (const float*)d_in[20];
    const float* updU_gate_w = (const float*)d_in[21];
    const float* updU_gate_b = (const float*)d_in[22];
    const float* updU_cand_w = (const float*)d_in[23];
    const float* updL_gate_w = (const float*)d_in[24];
    const float* updL_gate_b = (const float*)d_in[25];
    const float* updL_cand_w = (const float*)d_in[26];
    float* out = (float*)d_out;

    dim3 grid(1024), block(128);
    hipLaunchKernelGGL(half_graph_fused, grid, block, 0, stream,
        h_fea, xh_u, xh_l, xf, xp, pdp_w1, pdp_w2, att_w, att_b,
        compU_att_w, compU_att_b, compU_w, compL_att_w, compL_att_b, compL_w,
        decU_att_w, decU_att_b, decU_w, decL_att_w, decL_att_b, decL_w,
        updU_gate_w, updU_gate_b, updU_cand_w, updL_gate_w, updL_gate_b, updL_cand_w,
        out);
}